// FlameTextureSpace_82154134438549
// MI455X (gfx1250) — compile-verified
//
#include <hip/hip_runtime.h>
#include <hip/hip_bf16.h>

#ifndef __has_builtin
#define __has_builtin(x) 0
#endif

#define TEXN 256
#define TT   65536      // 256*256 texels
#define NB   16         // batch
#define NV   5023       // vertices
#define NP   60000      // valid pixels
#define SHW  512        // source image H=W
#define SIMG (3*SHW*SHW) // per-batch source image elems

#if __has_builtin(__builtin_amdgcn_global_load_async_to_lds_b32)
#define HAVE_ASYNC_LDS 1
#else
#define HAVE_ASYNC_LDS 0
#endif

typedef __attribute__((address_space(1))) int gas_int;
typedef __attribute__((address_space(3))) int las_int;

// ---- CDNA5 async global->LDS copy (ASYNCcnt-tracked), with portable fallback ----
__device__ __forceinline__ void async_b32(const void* g, void* l) {
#if HAVE_ASYNC_LDS
    __builtin_amdgcn_global_load_async_to_lds_b32(
        (gas_int*)(g), (las_int*)(l), 0, 0);
#else
    *reinterpret_cast<unsigned int*>(l) = *reinterpret_cast<const unsigned int*>(g);
#endif
}

__device__ __forceinline__ void wait_async0() {
#if HAVE_ASYNC_LDS
#if __has_builtin(__builtin_amdgcn_s_wait_asynccnt)
    __builtin_amdgcn_s_wait_asynccnt(0);
#else
    asm volatile("s_wait_asynccnt 0" ::: "memory");
#endif
#endif
}

// ---------------------------------------------------------------------------
// Pass 1: tail fill ONLY for texels t in [NP, TT) which keep the zero-grid
// default: gx = gy = 255.5 -> 4-corner average of the image center; mask = 0.
// (The p->texel map is the identity, so every t < NP is overwritten by the
// scatter pass — filling them here would waste ~45% of the store traffic.)
// ---------------------------------------------------------------------------
__global__ void flame_tail_kernel(const float* __restrict__ src,
                                  float* __restrict__ out_img,
                                  float* __restrict__ out_mask)
{
    const int t = NP + blockIdx.x * blockDim.x + threadIdx.x; // NP..TT-1
    if (t >= TT) return;
    const int b = blockIdx.y;                                 // batch (uniform)
    const float* pb = src + (size_t)b * SIMG;
#pragma unroll
    for (int c = 0; c < 3; ++c) {
        const float* pc = pb + c * (SHW * SHW);
        const float v = 0.25f * (pc[255 * SHW + 255] + pc[255 * SHW + 256] +
                                 pc[256 * SHW + 255] + pc[256 * SHW + 256]);
        __builtin_nontemporal_store(v, &out_img[((size_t)(b * 3 + c) << 16) + t]);
    }
    __builtin_nontemporal_store(0.0f, &out_mask[((size_t)b << 16) + t]);
}

// ---------------------------------------------------------------------------
// Pass 2: scatter. Each block stages a 256-pixel tile of the shared tables
// (faces/bary/x/y) into LDS via async global->LDS ops, then reuses the tile
// for 4 batches: barycentric blend -> camera projection -> bilinear
// zero-padded sample of source_img -> NT scatter stores.
// ---------------------------------------------------------------------------
__global__ void flame_scatter_kernel(const float* __restrict__ src,
                                     const float* __restrict__ mesh,
                                     const float* __restrict__ nrm,
                                     const float* __restrict__ cam,
                                     const int*   __restrict__ faces,
                                     const float* __restrict__ bc,
                                     const int*   __restrict__ xc,
                                     const int*   __restrict__ yc,
                                     float* __restrict__ out_img,
                                     float* __restrict__ out_mask)
{
    __shared__ int   sFace[256 * 3];
    __shared__ float sBc[256 * 3];
    __shared__ int   sX[256];
    __shared__ int   sY[256];

    const int tid = threadIdx.x;
    const int p0  = blockIdx.x * 256;

    // ---- stage tile (clamped so the last partial tile stays in-bounds and
    //      EXEC remains all-ones for the async ops) ----
    const int base3 = p0 * 3;
#pragma unroll
    for (int k = 0; k < 3; ++k) {
        const int j  = k * 256 + tid;
        const int gi = min(base3 + j, NP * 3 - 1);
        async_b32(faces + gi, &sFace[j]);
        async_b32(bc    + gi, &sBc[j]);
    }
    {
        const int gi = min(p0 + tid, NP - 1);
        async_b32(xc + gi, &sX[tid]);
        async_b32(yc + gi, &sY[tid]);
    }
    wait_async0();
    __syncthreads();

    const int  p    = p0 + tid;
    const bool live = (p < NP);

    const int   i0 = sFace[tid * 3 + 0];
    const int   i1 = sFace[tid * 3 + 1];
    const int   i2 = sFace[tid * 3 + 2];
    const float w0 = sBc[tid * 3 + 0];
    const float w1 = sBc[tid * 3 + 1];
    const float w2 = sBc[tid * 3 + 2];
    const int   xx = sX[tid];
    const int   yy = sY[tid];

    for (int ib = 0; ib < 4; ++ib) {
        const int b = blockIdx.y * 4 + ib;          // wave-uniform -> scalar loads
        const float f  = cam[b * 3 + 0];
        const float cx = cam[b * 3 + 1];
        const float cy = cam[b * 3 + 2];

        const float* mb = mesh + (size_t)b * NV * 3;
        const float* nb = nrm  + (size_t)b * NV * 3;

        // barycentric blend (only x,y of position and z of normal are needed)
        const float px = w0 * mb[i0 * 3 + 0] + w1 * mb[i1 * 3 + 0] + w2 * mb[i2 * 3 + 0];
        const float py = w0 * mb[i0 * 3 + 1] + w1 * mb[i1 * 3 + 1] + w2 * mb[i2 * 3 + 1];
        const float nz = w0 * nb[i0 * 3 + 2] + w1 * nb[i1 * 3 + 2] + w2 * nb[i2 * 3 + 2];

        // projection; grid y is negated
        const float gxn = f * (px + cx);
        const float gyn = -(f * (py + cy));

        // grid_sample, align_corners=False: gx = (g+1)*W/2 - 0.5
        const float gx = (gxn + 1.0f) * (0.5f * SHW) - 0.5f;
        const float gy = (gyn + 1.0f) * (0.5f * SHW) - 0.5f;

        const float x0f = floorf(gx), y0f = floorf(gy);
        const int   x0 = (int)x0f,    y0 = (int)y0f;
        const int   x1 = x0 + 1,      y1 = y0 + 1;
        const float wx1 = gx - x0f,   wy1 = gy - y0f;
        const float wx0 = 1.0f - wx1, wy0 = 1.0f - wy1;

        const bool vx0 = (x0 >= 0) && (x0 < SHW);
        const bool vx1 = (x1 >= 0) && (x1 < SHW);
        const bool vy0 = (y0 >= 0) && (y0 < SHW);
        const bool vy1 = (y1 >= 0) && (y1 < SHW);

        const int x0c = min(max(x0, 0), SHW - 1);
        const int x1c = min(max(x1, 0), SHW - 1);
        const int y0c = min(max(y0, 0), SHW - 1);
        const int y1c = min(max(y1, 0), SHW - 1);

        const float w00 = wx0 * wy0 * ((vx0 && vy0) ? 1.0f : 0.0f);
        const float w01 = wx1 * wy0 * ((vx1 && vy0) ? 1.0f : 0.0f);
        const float w10 = wx0 * wy1 * ((vx0 && vy1) ? 1.0f : 0.0f);
        const float w11 = wx1 * wy1 * ((vx1 && vy1) ? 1.0f : 0.0f);

        const int o00 = y0c * SHW + x0c;
        const int o01 = y0c * SHW + x1c;
        const int o10 = y1c * SHW + x0c;
        const int o11 = y1c * SHW + x1c;

        const float* pb = src + (size_t)b * SIMG;
#pragma unroll
        for (int c = 0; c < 3; ++c) {
            const float* pc = pb + c * (SHW * SHW);
            const float val = pc[o00] * w00 + pc[o01] * w01 +
                              pc[o10] * w10 + pc[o11] * w11;
            if (live)
                __builtin_nontemporal_store(
                    val, &out_img[((size_t)(b * 3 + c) << 16) + yy * TEXN + xx]);
        }
        if (live)
            __builtin_nontemporal_store(
                (nz < 0.0f) ? 1.0f : 0.0f,
                &out_mask[((size_t)b << 16) + yy * TEXN + xx]);
    }
}

extern "C" void kernel_launch(void* const* d_in, const int* in_sizes, int n_in,
                              void* d_out, int out_size, void* d_ws, size_t ws_size,
                              hipStream_t stream) {
    const float* src   = (const float*)d_in[0];  // [16,3,512,512]
    const float* mesh  = (const float*)d_in[1];  // [16,5023,3]
    const float* nrm   = (const float*)d_in[2];  // [16,5023,3]
    const float* cam   = (const float*)d_in[3];  // [16,3]
    const int*   faces = (const int*)d_in[4];    // [60000,3]
    const float* bc    = (const float*)d_in[5];  // [60000,3]
    const int*   xc    = (const int*)d_in[6];    // [60000]
    const int*   yc    = (const int*)d_in[7];    // [60000]

    float* out_img  = (float*)d_out;                       // [16,3,256,256]
    float* out_mask = out_img + (size_t)NB * 3 * TT;       // [16,1,256,256]

    dim3 gA((TT - NP + 255) / 256, NB);                    // tail texels only
    flame_tail_kernel<<<gA, 256, 0, stream>>>(src, out_img, out_mask);

    dim3 gB((NP + 255) / 256, NB / 4);
    flame_scatter_kernel<<<gB, 256, 0, stream>>>(src, mesh, nrm, cam, faces, bc,
                                                 xc, yc, out_img, out_mask);
}